// SparseGATLayer_46720654246366
// MI455X (gfx1250) — compile-verified
//
#include <hip/hip_runtime.h>
#include <math.h>

// ---------------------------------------------------------------------------
// Sparse GAT layer for MI455X (gfx1250, wave32)
//   B=4, N=20000, K=16, IN_DIM=256, H=4, C_H=64
// Phase 1: h = x @ W.T via v_wmma_f32_16x16x4_f32   (h -> workspace, 81.9MB,
//          stays resident in the 192MB L2 for phase 2's gathers)
// Phase 2: per-node gather of 16 neighbor rows, online softmax over K per
//          head, weighted aggregation, mean over heads. Indices preloaded as
//          int4 so the unrolled gather loop can keep max loads in flight.
// ---------------------------------------------------------------------------

typedef float v2f __attribute__((ext_vector_type(2)));
typedef float v8f __attribute__((ext_vector_type(8)));

#define GAT_B 4
#define GAT_N 20000
#define GAT_K 16
#define GAT_D 256     // IN_DIM == H*C_H
#define GAT_H 4
#define GAT_CH 64

// ---------------------------------------------------------------------------
// Phase 1: one wave computes a 16(M) x 64(N) tile of h = x @ W^T.
//   A tile (16x4 f32):  lane<16 -> {A[l][k], A[l][k+1]}, lane>=16 -> k+2,k+3
//   B tile (4x16 f32):  B[kk][o] = W[o][kk]  (same float2 addressing on W rows)
//   D (16x16 f32, 8 VGPRs): vgpr r, lane<16 -> (M=r, N=lane); lane>=16 -> M=r+8
// ---------------------------------------------------------------------------
__global__ __launch_bounds__(128)
void gat_gemm_wmma(const float* __restrict__ x,
                   const float* __restrict__ W,
                   float* __restrict__ h)
{
    const int lane = threadIdx.x & 31;
    const int wave = threadIdx.x >> 5;        // 0..3 : which 64-col group (head)
    const int m0   = blockIdx.x * 16;         // 5000 blocks
    const int n0   = wave * 64;
    const int half = lane >> 4;               // 0 | 1
    const int l16  = lane & 15;

    const float* arow = x + (size_t)(m0 + l16) * GAT_D;
    const float* b0r  = W + (size_t)(n0 +  0 + l16) * GAT_D;
    const float* b1r  = W + (size_t)(n0 + 16 + l16) * GAT_D;
    const float* b2r  = W + (size_t)(n0 + 32 + l16) * GAT_D;
    const float* b3r  = W + (size_t)(n0 + 48 + l16) * GAT_D;

    v8f acc0 = {}; v8f acc1 = {}; v8f acc2 = {}; v8f acc3 = {};

    #pragma unroll 4
    for (int k = 0; k < GAT_D; k += 4) {
        const int kk = k + 2 * half;
        v2f a  = *(const v2f*)(arow + kk);
        v2f b0 = *(const v2f*)(b0r + kk);
        v2f b1 = *(const v2f*)(b1r + kk);
        v2f b2 = *(const v2f*)(b2r + kk);
        v2f b3 = *(const v2f*)(b3r + kk);
        acc0 = __builtin_amdgcn_wmma_f32_16x16x4_f32(false, a, false, b0, (short)0, acc0, false, false);
        acc1 = __builtin_amdgcn_wmma_f32_16x16x4_f32(false, a, false, b1, (short)0, acc1, false, false);
        acc2 = __builtin_amdgcn_wmma_f32_16x16x4_f32(false, a, false, b2, (short)0, acc2, false, false);
        acc3 = __builtin_amdgcn_wmma_f32_16x16x4_f32(false, a, false, b3, (short)0, acc3, false, false);
    }

    #pragma unroll
    for (int r = 0; r < 8; ++r) {
        float* hrow = h + (size_t)(m0 + r + 8 * half) * GAT_D + n0 + l16;
        hrow[ 0] = acc0[r];
        hrow[16] = acc1[r];
        hrow[32] = acc2[r];
        hrow[48] = acc3[r];
    }
}

// ---------------------------------------------------------------------------
// Phase 2: one wave per node p = b*N + n.
//   lane l owns channels [8l, 8l+8)  => head = l>>3, ch-in-head = 8*(l&7)
// ---------------------------------------------------------------------------
__device__ __forceinline__ float dot4f(float4 a, float4 b) {
    return fmaf(a.x, b.x, fmaf(a.y, b.y, fmaf(a.z, b.z, a.w * b.w)));
}

__global__ __launch_bounds__(256)
void gat_gather_softmax(const float* __restrict__ h,
                        const int*   __restrict__ nbr,
                        const float* __restrict__ attn,
                        float*       __restrict__ out)
{
    const int lane = threadIdx.x & 31;
    const int wid  = threadIdx.x >> 5;
    const int p    = blockIdx.x * 8 + wid;     // node id, 0..B*N-1 (exact)
    const int b    = p / GAT_N;
    const int c0   = lane * 8;                 // global channel base 0..248
    const int head = lane >> 3;
    const int ch   = c0 & (GAT_CH - 1);

    // All 16 neighbor indices up-front (4x global_load_b128, wave-uniform):
    // removes the idx-load -> gather-load dependency inside the hot loop.
    const int4* idxv = (const int4*)(nbr + (size_t)p * GAT_K);
    int4 i0 = idxv[0];
    int4 i1 = idxv[1];
    int4 i2 = idxv[2];
    int4 i3 = idxv[3];
    int nbs[GAT_K] = { i0.x, i0.y, i0.z, i0.w,
                       i1.x, i1.y, i1.z, i1.w,
                       i2.x, i2.y, i2.z, i2.w,
                       i3.x, i3.y, i3.z, i3.w };

    // per-lane slices of attn vectors
    const float* ap = attn + head * (2 * GAT_CH) + ch;
    float4 as0 = *(const float4*)(ap);
    float4 as1 = *(const float4*)(ap + 4);
    float4 ad0 = *(const float4*)(ap + GAT_CH);
    float4 ad1 = *(const float4*)(ap + GAT_CH + 4);

    // s_src from own row
    const float* hp = h + (size_t)p * GAT_D + c0;
    float4 hi0 = *(const float4*)(hp);
    float4 hi1 = *(const float4*)(hp + 4);
    float ssrc = dot4f(hi0, as0) + dot4f(hi1, as1);
    ssrc += __shfl_xor(ssrc, 1, 32);
    ssrc += __shfl_xor(ssrc, 2, 32);
    ssrc += __shfl_xor(ssrc, 4, 32);           // all 8 lanes of head group hold s_src[h]

    const float* hbase = h + (size_t)b * GAT_N * GAT_D + c0;

    // online softmax over the 16 neighbors (single pass, no hj retention)
    float m = -3.402823466e+38f;
    float s = 0.0f;
    float a0 = 0.f, a1 = 0.f, a2 = 0.f, a3 = 0.f;
    float a4 = 0.f, a5 = 0.f, a6 = 0.f, a7 = 0.f;

    #pragma unroll
    for (int k = 0; k < GAT_K; ++k) {
        const float* jp = hbase + (size_t)nbs[k] * GAT_D;
        float4 j0 = *(const float4*)(jp);
        float4 j1 = *(const float4*)(jp + 4);

        float sd = dot4f(j0, ad0) + dot4f(j1, ad1);
        sd += __shfl_xor(sd, 1, 32);
        sd += __shfl_xor(sd, 2, 32);
        sd += __shfl_xor(sd, 4, 32);           // s_dst[k][h] in all 8 group lanes

        const float e  = (ssrc + sd) * 0.125f; // * 1/sqrt(C_H)
        const float mn = fmaxf(m, e);
        const float sc = __expf(m - mn);       // 0 on first iteration
        const float w  = __expf(e - mn);
        s = s * sc + w;
        m = mn;
        a0 = a0 * sc + w * j0.x;
        a1 = a1 * sc + w * j0.y;
        a2 = a2 * sc + w * j0.z;
        a3 = a3 * sc + w * j0.w;
        a4 = a4 * sc + w * j1.x;
        a5 = a5 * sc + w * j1.y;
        a6 = a6 * sc + w * j1.z;
        a7 = a7 * sc + w * j1.w;
    }

    const float inv = 1.0f / s;                // per-head softmax denominator
    float v[8] = { a0 * inv, a1 * inv, a2 * inv, a3 * inv,
                   a4 * inv, a5 * inv, a6 * inv, a7 * inv };

    // mean over the 4 heads: lanes l, l^8, l^16, l^24 share the same ch
    #pragma unroll
    for (int j = 0; j < 8; ++j) {
        float t = v[j];
        t += __shfl_xor(t, 8, 32);
        t += __shfl_xor(t, 16, 32);
        v[j] = t * 0.25f;
    }

    if (lane < 8) {
        float* op = out + (size_t)p * GAT_CH + lane * 8;
        *(float4*)(op)     = make_float4(v[0], v[1], v[2], v[3]);
        *(float4*)(op + 4) = make_float4(v[4], v[5], v[6], v[7]);
    }
}

// ---------------------------------------------------------------------------
extern "C" void kernel_launch(void* const* d_in, const int* in_sizes, int n_in,
                              void* d_out, int out_size, void* d_ws, size_t ws_size,
                              hipStream_t stream)
{
    const float* x    = (const float*)d_in[0];   // (B,N,256) f32
    const int*   idx  = (const int*)  d_in[1];   // (B,N,16)  i32
    const float* W    = (const float*)d_in[2];   // (256,256) f32
    const float* attn = (const float*)d_in[3];   // (4,128)   f32
    float* out = (float*)d_out;                  // (B,N,64)  f32
    float* h   = (float*)d_ws;                   // (B*N,256) f32 = 81.92 MB

    // Phase 1: h = x @ W.T   -- 5000 m-tiles, 4 waves/block (one head each)
    gat_gemm_wmma<<<(GAT_B * GAT_N) / 16, 128, 0, stream>>>(x, W, h);

    // Phase 2: gather + softmax + aggregate -- 8 nodes per 256-thread block
    gat_gather_softmax<<<(GAT_B * GAT_N) / 8, 256, 0, stream>>>(h, idx, attn, out);
}